// SocialAttention_2422361555838
// MI455X (gfx1250) — compile-verified
//
#include <hip/hip_runtime.h>
#include <hip/hip_bf16.h>

#define NN 8192
#define DIM 128
#define KCH 128          // K-chunk for LDS-staged GEMM
#define APAD 132         // padded alpha-tile row stride (floats): banks (4l+ka)%64 all distinct

typedef __attribute__((ext_vector_type(2))) float v2f;
typedef __attribute__((ext_vector_type(8))) float v8f;

// ---------------------------------------------------------------------------
// Kernel 1: Wh = h @ W^T   ([N,128] = [N,128] @ [128,128]^T), exact fp32 WMMA.
// One wave computes one 16x16 tile. Block = 8 waves = one 16-row strip across
// all 8 N-tiles. grid = N/16 blocks. (Tiny: 0.27 GFLOP, loads are b64-clean.)
// WMMA f32 16x16x4 fragment layout (ISA 7.12.2):
//   A (16x4): lanes 0-15 -> M=lane, regs {K=0,K=1}; lanes 16-31 -> {K=2,K=3}
//   B (4x16): lanes 0-15 -> N=lane, regs {K=0,K=1}; lanes 16-31 -> {K=2,K=3}
//   D (16x16): reg r: lanes 0-15 -> M=r, lanes 16-31 -> M=8+r; N = lane&15
// ---------------------------------------------------------------------------
__global__ __launch_bounds__(256) void wh_wmma_kernel(const float* __restrict__ h,
                                                      const float* __restrict__ W,
                                                      float* __restrict__ Wh) {
    const int wave = threadIdx.x >> 5;
    const int lane = threadIdx.x & 31;
    const int m0   = blockIdx.x * 16;
    const int n0   = wave * 16;
    const bool hi  = lane >= 16;
    const int l    = lane & 15;
    const int m    = m0 + l;
    const int n    = n0 + l;

    v8f acc = {};
    #pragma unroll 4
    for (int k0 = 0; k0 < DIM; k0 += 4) {
        const int ka = k0 + (hi ? 2 : 0);
        v2f a, b;
        a.x = h[m * DIM + ka];          // contiguous pair -> global_load_b64
        a.y = h[m * DIM + ka + 1];
        b.x = W[n * DIM + ka];          // B[k][n] = W[n][k], also contiguous
        b.y = W[n * DIM + ka + 1];
        acc = __builtin_amdgcn_wmma_f32_16x16x4_f32(
            false, a, false, b, (short)0, acc, false, false);
    }

    const int mBase = m0 + (hi ? 8 : 0);
    #pragma unroll
    for (int r = 0; r < 8; ++r)
        Wh[(mBase + r) * DIM + n] = acc[r];
}

// ---------------------------------------------------------------------------
// Kernel 2: a1[i] = Wh[i,:] . a[0:128],  a2[i] = Wh[i,:] . a[128:256]
// ---------------------------------------------------------------------------
__global__ __launch_bounds__(256) void a12_kernel(const float* __restrict__ Wh,
                                                  const float* __restrict__ a,
                                                  float* __restrict__ a1,
                                                  float* __restrict__ a2) {
    const int i = blockIdx.x * blockDim.x + threadIdx.x;
    float s1 = 0.f, s2 = 0.f;
    #pragma unroll 8
    for (int k = 0; k < DIM; ++k) {
        const float w = Wh[i * DIM + k];
        s1 += w * a[k];
        s2 += w * a[DIM + k];
    }
    a1[i] = s1;
    a2[i] = s2;
}

// ---------------------------------------------------------------------------
// Kernel 3: masked softmax row -> alpha.  One 256-thread block per row.
// adj row read exactly once (mask bits live in one register per thread);
// rows with zero neighbors output alpha=0 (matches reference no_neighbor
// zeroing, and makes out-rows zero for free in kernel 4).
// ---------------------------------------------------------------------------
__global__ __launch_bounds__(256) void softmax_kernel(const int* __restrict__ adj,
                                                      const float* __restrict__ a1,
                                                      const float* __restrict__ a2,
                                                      float* __restrict__ alpha) {
    __shared__ float p_s[NN];      // 32 KB: un-normalized probabilities
    __shared__ float redf[256];
    __shared__ int   redi[256];

    const int i   = blockIdx.x;
    const int tid = threadIdx.x;
    const float a1i = a1[i];
    const size_t rowBase = (size_t)i * NN;

    unsigned mask = 0u;
    float lmax = -3.0e38f;
    int cnt = 0;
    for (int c = 0; c < 32; ++c) {
        const int j = c * 256 + tid;                 // coalesced per c
        if (adj[rowBase + j] != 0) {
            mask |= (1u << c);
            const float s = a1i + a2[j];
            const float e = s > 0.f ? s : 0.2f * s;
            lmax = fmaxf(lmax, e);
            ++cnt;
        }
    }
    redf[tid] = lmax; redi[tid] = cnt;
    __syncthreads();
    for (int off = 128; off > 0; off >>= 1) {
        if (tid < off) {
            redf[tid] = fmaxf(redf[tid], redf[tid + off]);
            redi[tid] += redi[tid + off];
        }
        __syncthreads();
    }
    const float rmax  = redf[0];
    const int   total = redi[0];
    __syncthreads();

    float lsum = 0.f;
    for (int c = 0; c < 32; ++c) {
        const int j = c * 256 + tid;
        float p = 0.f;
        if (mask & (1u << c)) {
            const float s = a1i + a2[j];
            const float e = s > 0.f ? s : 0.2f * s;
            p = __expf(e - rmax);
            lsum += p;
        }
        p_s[j] = p;
    }
    redf[tid] = lsum;
    __syncthreads();
    for (int off = 128; off > 0; off >>= 1) {
        if (tid < off) redf[tid] += redf[tid + off];
        __syncthreads();
    }
    const float inv = (total > 0) ? (1.0f / redf[0]) : 0.0f;

    for (int c = 0; c < 32; ++c) {
        const int j = c * 256 + tid;
        alpha[rowBase + j] = p_s[j] * inv;           // coalesced store
    }
}

// ---------------------------------------------------------------------------
// Kernel 4: out = alpha @ Wh   ([N,128] = [N,N] @ [N,128]), fp32 WMMA with
// LDS staging.  Block = 16 output rows x all 128 cols (8 waves).  Per 128-K
// chunk: stage alpha[16x128] (padded, conflict-free ds_load_b64 A-fragments,
// shared by all 8 waves) and Wh[128x128] via dense global_load_b128, then
// 32 WMMAs per wave from LDS.  LDS ~73 KB -> 4 blocks/WGP.
// ---------------------------------------------------------------------------
__global__ __launch_bounds__(256) void out_wmma_kernel(const float* __restrict__ alpha,
                                                       const float* __restrict__ Wh,
                                                       float* __restrict__ out) {
    __shared__ float as_s[16 * APAD];    // alpha chunk, 528B row stride (16B-aligned)
    __shared__ float wh_s[KCH * DIM];    // Wh chunk, 64 KB

    const int tid  = threadIdx.x;
    const int wave = tid >> 5;
    const int lane = tid & 31;
    const int m0   = blockIdx.x * 16;
    const int n0   = wave * 16;
    const bool hi  = lane >= 16;
    const int l    = lane & 15;
    const int n    = n0 + l;

    v8f acc = {};
    for (int k0 = 0; k0 < NN; k0 += KCH) {
        // --- stage alpha tile [16 x KCH]: 512 float4, 2 per thread ---
        #pragma unroll
        for (int p = 0; p < 2; ++p) {
            const int idx = p * 256 + tid;           // float4 index
            const int r = idx >> 5, c4 = idx & 31;   // 32 float4 per row
            const float4 v = *(const float4*)(alpha + (size_t)(m0 + r) * NN + k0 + c4 * 4);
            *(float4*)(&as_s[r * APAD + c4 * 4]) = v;
        }
        // --- stage Wh tile [KCH x 128]: 4096 float4, 16 per thread ---
        #pragma unroll
        for (int p = 0; p < 16; ++p) {
            const int idx = p * 256 + tid;
            const int r = idx >> 5, c4 = idx & 31;
            const float4 v = *(const float4*)(Wh + (size_t)(k0 + r) * DIM + c4 * 4);
            *(float4*)(&wh_s[r * DIM + c4 * 4]) = v;
        }
        // prefetch next chunk's alpha stream (HBM-resident, 256MB total)
        if (k0 + KCH < NN) {
            const int pr = tid >> 4, pc = (tid & 15) * 8;    // 16 rows x 16 probes
            __builtin_prefetch(alpha + (size_t)(m0 + pr) * NN + k0 + KCH + pc, 0, 1);
        }
        __syncthreads();

        #pragma unroll 4
        for (int kk = 0; kk < KCH; kk += 4) {
            const int ka = kk + (hi ? 2 : 0);        // even -> 8B-aligned LDS reads
            const v2f a = *(const v2f*)(&as_s[l * APAD + ka]);   // ds_load_b64
            v2f b;
            b.x = wh_s[ka * DIM + n];                // ds_load_2addr pair
            b.y = wh_s[(ka + 1) * DIM + n];
            acc = __builtin_amdgcn_wmma_f32_16x16x4_f32(
                false, a, false, b, (short)0, acc, false, false);
        }
        __syncthreads();
    }

    const int mBase = m0 + (hi ? 8 : 0);
    #pragma unroll
    for (int r = 0; r < 8; ++r)
        out[(size_t)(mBase + r) * DIM + n] = acc[r];
}

// ---------------------------------------------------------------------------
extern "C" void kernel_launch(void* const* d_in, const int* in_sizes, int n_in,
                              void* d_out, int out_size, void* d_ws, size_t ws_size,
                              hipStream_t stream) {
    (void)in_sizes; (void)n_in; (void)out_size; (void)ws_size;

    const float* h   = (const float*)d_in[0];
    const int*   adj = (const int*)d_in[1];
    const float* W   = (const float*)d_in[2];
    const float* a   = (const float*)d_in[3];

    float* out   = (float*)d_out;                    // [N, DIM]
    float* alpha = out + (size_t)NN * DIM;           // [N, N] (second output)

    float* Wh = (float*)d_ws;                        // [N, DIM] = 4 MB
    float* a1 = Wh + (size_t)NN * DIM;               // [N]
    float* a2 = a1 + NN;                             // [N]

    wh_wmma_kernel <<<NN / 16, 256, 0, stream>>>(h, W, Wh);
    a12_kernel     <<<NN / 256, 256, 0, stream>>>(Wh, a, a1, a2);
    softmax_kernel <<<NN, 256, 0, stream>>>(adj, a1, a2, alpha);
    out_wmma_kernel<<<NN / 16, 256, 0, stream>>>(alpha, Wh, out);
}